// HastGCN_72593537237178
// MI455X (gfx1250) — compile-verified
//
#include <hip/hip_runtime.h>
#include <math.h>

#define Bc   8
#define Tc   12
#define Nc   883
#define NRc  100
#define DINc 3
#define Hc   16
#define TMc  6
#define ESSc 16000
#define ERRc 1600

typedef __attribute__((ext_vector_type(2))) float v2f;
typedef __attribute__((ext_vector_type(8))) float v8f;

__device__ __forceinline__ float lrelu(float x) { return x >= 0.f ? x : 0.2f * x; }
// optimization barrier: force value into a VGPR so selects lower to v_cndmask
__device__ __forceinline__ void vreg(float& x) { asm volatile("" : "+v"(x)); }

// ---------------------------------------------------------------- utilities
__global__ void k_zero(float* __restrict__ p, long n) {
  long i = (long)blockIdx.x * 256 + threadIdx.x;
  if (i < n) p[i] = 0.f;
}

// ------------------------------------------- start conv + GAT-S projections
__global__ void k_start_proj(const float* __restrict__ data,
                             const float* __restrict__ w_start,
                             const float* __restrict__ b_start,
                             const float* __restrict__ wg,
                             const float* __restrict__ a1,
                             const float* __restrict__ a2,
                             float* __restrict__ h, float* __restrict__ alpha,
                             float* __restrict__ beta) {
  long idx = (long)blockIdx.x * 256 + threadIdx.x;
  long tot = (long)Bc * Tc * Nc;
  if (idx >= tot) return;
  const float* dp = data + idx * DINc;
  float d0 = dp[0], d1 = dp[1], d2 = dp[2];
  float x[Hc];
#pragma unroll
  for (int e = 0; e < Hc; ++e) {
    float v = d0 * w_start[e * DINc + 0] + d1 * w_start[e * DINc + 1] +
              d2 * w_start[e * DINc + 2] + b_start[e];
    x[e] = v > 0.f ? v : 0.f;
  }
  float av = 0.f, bv = 0.f;
  float* hp = h + idx * Hc;
#pragma unroll
  for (int e = 0; e < Hc; ++e) {
    float s = 0.f;
#pragma unroll
    for (int d = 0; d < Hc; ++d) s += x[d] * wg[d * Hc + e];
    hp[e] = s;
    av += s * a1[e];
    bv += s * a2[e];
  }
  alpha[idx] = av;
  beta[idx] = bv;
}

// generic 16->16 projection for the R graph
__global__ void k_proj(const float* __restrict__ in, const float* __restrict__ wg,
                       const float* __restrict__ a1, const float* __restrict__ a2,
                       float* __restrict__ h, float* __restrict__ alpha,
                       float* __restrict__ beta, long tot) {
  long idx = (long)blockIdx.x * 256 + threadIdx.x;
  if (idx >= tot) return;
  const float* ip = in + idx * Hc;
  float x[Hc];
#pragma unroll
  for (int d = 0; d < Hc; ++d) x[d] = ip[d];
  float av = 0.f, bv = 0.f;
  float* hp = h + idx * Hc;
#pragma unroll
  for (int e = 0; e < Hc; ++e) {
    float s = 0.f;
#pragma unroll
    for (int d = 0; d < Hc; ++d) s += x[d] * wg[d * Hc + e];
    hp[e] = s;
    av += s * a1[e];
    bv += s * a2[e];
  }
  alpha[idx] = av;
  beta[idx] = bv;
}

// ------------------------------------------------------- masked-softmax GAT
// one wave per (b*t, 16-row M-tile). Two-pass masked softmax with fp32 WMMA
// (16x16x4) accumulation of P @ h. Fully branchless K-loop: all loads
// unconditional, mask select forced to v_cndmask via VGPR barrier, masked
// probability is the per-row constant pmask = exp(-1e9 - m) (== 1 when the
// whole row is masked -> uniform softmax, == 0 otherwise), matching jnp.
__global__ void k_gat(const float* __restrict__ adj, int Nn,
                      const float* __restrict__ h, const float* __restrict__ alpha,
                      const float* __restrict__ beta, float* __restrict__ out) {
  int ntiles = (Nn + 15) >> 4;
  int bt = blockIdx.x / ntiles;
  int tile = blockIdx.x - bt * ntiles;
  int base = tile << 4;
  int L = threadIdx.x;
  int lane = L & 15;
  bool hih = L >= 16;
  int row = base + lane;
  int rowc = row < Nn ? row : 0;  // clamped: invalid rows compute garbage rows
  const float* hB = h + (long)bt * Nn * Hc;
  const float* aB = alpha + (long)bt * Nn;
  const float* bB = beta + (long)bt * Nn;
  float arow = aB[rowc];
  const float* adjrow = adj + (long)rowc * Nn;

  // ---- pass 1: masked row max; uniform trip count, branchless body
  float m = -1e9f;
  int halfn = (Nn + 1) >> 1;
  int js = hih ? halfn : 0;
  for (int i = 0; i < halfn; ++i) {
    int j = js + i;
    int jc = j < Nn ? j : 0;
    float adjv = adjrow[jc];
    float bv = bB[jc];
    float e = lrelu(arow + bv);
    vreg(e);
    float v = (j < Nn && adjv != 0.f) ? e : -1e9f;
    m = fmaxf(m, v);
  }
  m = fmaxf(m, __shfl_xor(m, 16));
  float pmask = __expf(-1e9f - m);  // 1 if row fully masked, else 0

  // ---- pass 2: WMMA accumulate of exp(e-m) @ h, tracking row sums
  v8f acc = {};
  float lsum = 0.f;
  int koff = hih ? 2 : 0;
  int nfull = Nn & ~3;
  for (int j0 = 0; j0 < nfull; j0 += 4) {
    if ((j0 & 63) == 0)  // prefetch upcoming h rows (global_prefetch_b8)
      __builtin_prefetch(hB + (long)(j0 + 64) * Hc + lane, 0, 1);
    int ja = j0 + koff;
    float adjA = adjrow[ja];
    float adjB = adjrow[ja + 1];
    float bA = bB[ja];
    float bBv = bB[ja + 1];
    float hA = hB[ja * Hc + lane];
    float hBv = hB[(ja + 1) * Hc + lane];
    float pea = __expf(lrelu(arow + bA) - m);
    float peb = __expf(lrelu(arow + bBv) - m);
    vreg(pea);
    vreg(peb);
    float pa = (adjA != 0.f) ? pea : pmask;
    float pb = (adjB != 0.f) ? peb : pmask;
    v2f A;
    A[0] = pa;
    A[1] = pb;
    v2f Bf;
    Bf[0] = hA;
    Bf[1] = hBv;
    acc = __builtin_amdgcn_wmma_f32_16x16x4_f32(false, A, false, Bf, (short)0,
                                                acc, false, false);
    lsum += pa + pb;
  }
  if (nfull < Nn) {  // peeled K-tail, executed once
    int ja = nfull + koff, jb = ja + 1;
    float pa = 0.f, pb = 0.f, hA = 0.f, hBv = 0.f;
    if (ja < Nn) {
      float pe = __expf(lrelu(arow + bB[ja]) - m);
      vreg(pe);
      pa = (adjrow[ja] != 0.f) ? pe : pmask;
      hA = hB[ja * Hc + lane];
    }
    if (jb < Nn) {
      float pe = __expf(lrelu(arow + bB[jb]) - m);
      vreg(pe);
      pb = (adjrow[jb] != 0.f) ? pe : pmask;
      hBv = hB[jb * Hc + lane];
    }
    v2f A;
    A[0] = pa;
    A[1] = pb;
    v2f Bf;
    Bf[0] = hA;
    Bf[1] = hBv;
    acc = __builtin_amdgcn_wmma_f32_16x16x4_f32(false, A, false, Bf, (short)0,
                                                acc, false, false);
    lsum += pa + pb;
  }
  lsum += __shfl_xor(lsum, 16);  // lane i (and i+16) now holds l for row i

  float* oB = out + (long)bt * Nn * Hc;
#pragma unroll
  for (int v = 0; v < 8; ++v) {
    int M = v + (hih ? 8 : 0);
    float l = __shfl(lsum, M);
    int orow = base + M;
    if (orow < Nn) oB[orow * Hc + lane] = acc[v] / l;
  }
}

// ------------------------------------------ r = relu( scaled_relu(x) @ adj_sr )
__global__ void k_rmat(const float* __restrict__ adj_sr, const float* __restrict__ s,
                       float* __restrict__ r) {
  const int ntiles = (NRc + 15) >> 4;  // 7
  int bt = blockIdx.x / ntiles;
  int tile = blockIdx.x - bt * ntiles;
  int base = tile << 4;
  int L = threadIdx.x;
  int lane = L & 15;
  bool hih = L >= 16;
  int rr = base + lane;
  int rrc = rr < NRc ? rr : 0;  // clamped A row
  const float* sB = s + (long)bt * Nc * Hc;
  v8f acc = {};
  int koff = hih ? 2 : 0;
  const int nfull = Nc & ~3;  // 880
  for (int n0 = 0; n0 < nfull; n0 += 4) {
    if ((n0 & 63) == 0)
      __builtin_prefetch(sB + (long)(n0 + 64) * Hc + lane, 0, 1);
    int na = n0 + koff;
    v2f A, Bf;
    A[0] = adj_sr[na * NRc + rrc];
    A[1] = adj_sr[(na + 1) * NRc + rrc];
    Bf[0] = sB[na * Hc + lane];
    Bf[1] = sB[(na + 1) * Hc + lane];
    acc = __builtin_amdgcn_wmma_f32_16x16x4_f32(false, A, false, Bf, (short)0,
                                                acc, false, false);
  }
  {  // peeled tail (Nc % 4 == 3)
    int na = nfull + koff, nb = na + 1;
    v2f A, Bf;
    A[0] = (na < Nc) ? adj_sr[na * NRc + rrc] : 0.f;
    A[1] = (nb < Nc) ? adj_sr[nb * NRc + rrc] : 0.f;
    Bf[0] = (na < Nc) ? sB[na * Hc + lane] : 0.f;
    Bf[1] = (nb < Nc) ? sB[nb * Hc + lane] : 0.f;
    acc = __builtin_amdgcn_wmma_f32_16x16x4_f32(false, A, false, Bf, (short)0,
                                                acc, false, false);
  }
  float* rB = r + (long)bt * NRc * Hc;
#pragma unroll
  for (int v = 0; v < 8; ++v) {
    int M = v + (hih ? 8 : 0);
    int orow = base + M;
    if (orow < NRc) rB[orow * Hc + lane] = fmaxf(acc[v], 0.f);
  }
}

// ------------------------------------------------------------ ChebConv parts
__global__ void k_deg(const int* __restrict__ eind, int E, float* __restrict__ deg) {
  int e = blockIdx.x * 256 + threadIdx.x;
  if (e >= E) return;
  atomicAdd(&deg[eind[e]], 1.0f);
}

__global__ void k_norm(const int* __restrict__ eind, int E,
                       const float* __restrict__ deg, float* __restrict__ norm) {
  int e = blockIdx.x * 256 + threadIdx.x;
  if (e >= E) return;
  int s = eind[e], d = eind[E + e];
  float ds = deg[s], dd = deg[d];
  float a = ds > 0.f ? rsqrtf(ds) : 0.f;
  float b = dd > 0.f ? rsqrtf(dd) : 0.f;
  norm[e] = -a * b;
}

__global__ void k_prop(const int* __restrict__ eind, const float* __restrict__ norm,
                       int E, int Nn, const float* __restrict__ z,
                       float* __restrict__ out) {
  long idx = (long)blockIdx.x * 256 + threadIdx.x;
  long tot = (long)E * Bc * Tc;
  if (idx >= tot) return;
  int e = (int)(idx % E);
  int bt = (int)(idx / E);
  int s = eind[e], d = eind[E + e];
  float nm = norm[e];
  const float* zp = z + ((long)bt * Nn + s) * Hc;
  float* op = out + ((long)bt * Nn + d) * Hc;
#pragma unroll
  for (int k = 0; k < Hc; ++k) atomicAdd(&op[k], zp[k] * nm);
}

// out = tx0@W0 + tx1@W1 + (2*P - tx0)@W2 (+bias). May write in place.
__global__ void k_cheb(const float* __restrict__ tx0, const float* __restrict__ tx1,
                       const float* __restrict__ P, const float* __restrict__ wc,
                       const float* __restrict__ bias, float* __restrict__ out,
                       long tot) {
  long idx = (long)blockIdx.x * 256 + threadIdx.x;
  if (idx >= tot) return;
  float x0[Hc], x1[Hc], x2[Hc], res[Hc];
  const float* p0 = tx0 + idx * Hc;
  const float* p1 = tx1 + idx * Hc;
  const float* p2 = P + idx * Hc;
#pragma unroll
  for (int d = 0; d < Hc; ++d) {
    x0[d] = p0[d];
    x1[d] = p1[d];
    x2[d] = 2.f * p2[d] - x0[d];
  }
#pragma unroll
  for (int e = 0; e < Hc; ++e) {
    float s = bias ? bias[e] : 0.f;
#pragma unroll
    for (int d = 0; d < Hc; ++d)
      s += x0[d] * wc[d * Hc + e] + x1[d] * wc[256 + d * Hc + e] +
           x2[d] * wc[512 + d * Hc + e];
    res[e] = s;
  }
  float* op = out + idx * Hc;
#pragma unroll
  for (int e = 0; e < Hc; ++e) op[e] = res[e];
}

// ---------------------------------------------- temporal attention (6x16 tiles)
__global__ void k_ta(const float* __restrict__ x, const float* __restrict__ wq,
                     const float* __restrict__ wk, const float* __restrict__ wv,
                     float* __restrict__ out) {
  int idx = blockIdx.x;
  int half = idx & 1;
  idx >>= 1;
  int n = idx % Nc;
  int b = idx / Nc;
  int toff = half * TMc;
  __shared__ float xs[6][16], qs[6][16], ks[6][16], vs[6][16], ss[6][6];
  int tid = threadIdx.x;
  int t = tid >> 4;
  int d = tid & 15;
  xs[t][d] = x[(((long)b * Tc + toff + t) * Nc + n) * Hc + d];
  __syncthreads();
  float q = 0.f, k = 0.f, v = 0.f;
#pragma unroll
  for (int dd = 0; dd < Hc; ++dd) {
    float xv = xs[t][dd];
    q += xv * wq[dd * Hc + d];
    k += xv * wk[dd * Hc + d];
    v += xv * wv[dd * Hc + d];
  }
  qs[t][d] = q;
  ks[t][d] = k;
  vs[t][d] = v;
  __syncthreads();
  if (tid < 36) {
    int ti = tid / 6, tj = tid % 6;
    float s = 0.f;
#pragma unroll
    for (int dd = 0; dd < Hc; ++dd) s += qs[ti][dd] * ks[tj][dd];
    ss[ti][tj] = s * 0.25f;  // 1/sqrt(16)
  }
  __syncthreads();
  if (tid < 6) {
    float mx = ss[tid][0];
#pragma unroll
    for (int j = 1; j < 6; ++j) mx = fmaxf(mx, ss[tid][j]);
    float sum = 0.f;
#pragma unroll
    for (int j = 0; j < 6; ++j) {
      float e = __expf(ss[tid][j] - mx);
      ss[tid][j] = e;
      sum += e;
    }
    float inv = 1.f / sum;
#pragma unroll
    for (int j = 0; j < 6; ++j) ss[tid][j] *= inv;
  }
  __syncthreads();
  float o = 0.f;
#pragma unroll
  for (int tj = 0; tj < 6; ++tj) o += ss[t][tj] * vs[tj][d];
  out[(((long)b * Tc + toff + t) * Nc + n) * Hc + d] = o;
}

// ---------------------------------------------------------------- gate + scale
__global__ void k_gate(const float* __restrict__ ta, const float* __restrict__ gw_q,
                       const float* __restrict__ gw_f, const float* __restrict__ gw_n,
                       float* __restrict__ g) {
  int b = blockIdx.x;
  __shared__ float red[256];
  float part = 0.f;
  for (int n = threadIdx.x; n < Nc; n += 256) {
    float sn = 0.f;
#pragma unroll
    for (int t = 0; t < Tc - TMc; ++t) {
      const float* p = ta + (((long)b * Tc + TMc + t) * Nc + n) * Hc;
      float st = 0.f;
#pragma unroll
      for (int d = 0; d < Hc; ++d) st += gw_f[d] * p[d];
      sn += gw_q[t] * st;
    }
    part += gw_n[n] * sn;
  }
  red[threadIdx.x] = part;
  __syncthreads();
  for (int s = 128; s > 0; s >>= 1) {
    if (threadIdx.x < s) red[threadIdx.x] += red[threadIdx.x + s];
    __syncthreads();
  }
  if (threadIdx.x == 0) g[b] = 1.f / (1.f + __expf(-red[0]));
}

__global__ void k_scale(float* __restrict__ buf, const float* __restrict__ g) {
  long idx = (long)blockIdx.x * 256 + threadIdx.x;
  long tot = (long)Bc * Tc * Nc * Hc;
  if (idx >= tot) return;
  long nh = (long)Nc * Hc;
  int t = (int)((idx / nh) % Tc);
  int b = (int)(idx / (nh * Tc));
  float gb = g[b];
  float sc = (t < TMc) ? (1.f - gb) : gb;
  float v = buf[idx] * sc;
  buf[idx] = v > 0.f ? v : 0.f;
}

// ------------------------------------------------------------- final contraction
__global__ void k_out(const float* __restrict__ r, const float* __restrict__ w_out,
                      const float* __restrict__ b_out, float* __restrict__ out) {
  int b = blockIdx.x / DINc;
  int o = blockIdx.x % DINc;
  __shared__ float red[256];
  float part = 0.f;
  const int Ltot = Tc * NRc * Hc;  // 19200
  for (int i = threadIdx.x; i < Ltot; i += 256) {
    int t = i / (NRc * Hc);
    int rem = i - t * NRc * Hc;
    int rr = rem / Hc;
    int d = rem - rr * Hc;
    float rv = r[(((long)b * Tc + t) * NRc + rr) * Hc + d];
    float wv = w_out[((long)o * Hc + d) * (NRc * Tc) + t * NRc + rr];
    part += rv * wv;
  }
  red[threadIdx.x] = part;
  __syncthreads();
  for (int s = 128; s > 0; s >>= 1) {
    if (threadIdx.x < s) red[threadIdx.x] += red[threadIdx.x + s];
    __syncthreads();
  }
  if (threadIdx.x == 0) out[b * DINc + o] = red[0] + b_out[o];
}

// ============================================================== host launcher
extern "C" void kernel_launch(void* const* d_in, const int* in_sizes, int n_in,
                              void* d_out, int out_size, void* d_ws, size_t ws_size,
                              hipStream_t stream) {
  (void)in_sizes; (void)n_in; (void)out_size; (void)ws_size;
  const float* data    = (const float*)d_in[0];
  const float* W       = (const float*)d_in[1];
  const float* adj_sr  = (const float*)d_in[2];
  const float* adj_rr  = (const float*)d_in[3];
  const int*   eind_ss = (const int*)d_in[4];
  const int*   eind_rr = (const int*)d_in[5];
  const float* w_start = (const float*)d_in[6];
  const float* b_start = (const float*)d_in[7];
  const float* wg_s    = (const float*)d_in[8];
  const float* a1_s    = (const float*)d_in[9];
  const float* a2_s    = (const float*)d_in[10];
  const float* wc_s    = (const float*)d_in[11];
  const float* wq_t    = (const float*)d_in[12];
  const float* wk_t    = (const float*)d_in[13];
  const float* wv_t    = (const float*)d_in[14];
  const float* gw_q    = (const float*)d_in[15];
  const float* gw_f    = (const float*)d_in[16];
  const float* gw_n    = (const float*)d_in[17];
  const float* wg_r    = (const float*)d_in[18];
  const float* a1_r    = (const float*)d_in[19];
  const float* a2_r    = (const float*)d_in[20];
  const float* wc_r    = (const float*)d_in[21];
  const float* b_chebr = (const float*)d_in[22];
  const float* w_out   = (const float*)d_in[23];
  const float* b_out   = (const float*)d_in[24];
  float* outp = (float*)d_out;

  const long SZ_BIG = (long)Bc * Tc * Nc * Hc;   // 1,356,288
  const long SZ_BTN = (long)Bc * Tc * Nc;        // 84,768
  const long SZ_R   = (long)Bc * Tc * NRc * Hc;  // 153,600
  const long SZ_BTR = (long)Bc * Tc * NRc;       // 9,600
  float* ws   = (float*)d_ws;
  float* bufG = ws;                long off = SZ_BIG;
  float* bufH = ws + off;          off += SZ_BIG;
  float* bufP = ws + off;          off += SZ_BIG;
  float* alS  = ws + off;          off += SZ_BTN;
  float* beS  = ws + off;          off += SZ_BTN;
  float* degS = ws + off;          off += Nc;
  float* nrmS = ws + off;          off += ESSc;
  float* gbuf = ws + off;          off += Bc;
  float* bufR = ws + off;          off += SZ_R;
  float* hR   = ws + off;          off += SZ_R;
  float* alR  = ws + off;          off += SZ_BTR;
  float* beR  = ws + off;          off += SZ_BTR;
  float* gatR = ws + off;          off += SZ_R;
  float* tx1R = ws + off;          off += SZ_R;
  float* prR  = ws + off;          off += SZ_R;
  float* degR = ws + off;          off += NRc;
  float* nrmR = ws + off;          off += ERRc;

  auto blks = [](long n) { return (unsigned)((n + 255) / 256); };

  // 1) start conv + GAT-S projections -> bufH (h), alS, beS
  k_start_proj<<<blks(SZ_BTN), 256, 0, stream>>>(data, w_start, b_start, wg_s,
                                                 a1_s, a2_s, bufH, alS, beS);
  // Cheb-S degree/norm
  k_zero<<<blks(Nc), 256, 0, stream>>>(degS, Nc);
  k_deg<<<blks(ESSc), 256, 0, stream>>>(eind_ss, ESSc, degS);
  k_norm<<<blks(ESSc), 256, 0, stream>>>(eind_ss, ESSc, degS, nrmS);

  // 2) GAT-S (WMMA) -> bufG
  {
    int ntiles = (Nc + 15) >> 4;  // 56
    k_gat<<<(unsigned)(Bc * Tc * ntiles), 32, 0, stream>>>(W, Nc, bufH, alS, beS, bufG);
  }

  // 3) Cheb-S: tx0 = bufG; tx1 -> bufH; P = prop(tx1) -> bufP; combine -> bufG
  k_zero<<<blks(SZ_BIG), 256, 0, stream>>>(bufH, SZ_BIG);
  k_prop<<<blks((long)ESSc * Bc * Tc), 256, 0, stream>>>(eind_ss, nrmS, ESSc, Nc,
                                                         bufG, bufH);
  k_zero<<<blks(SZ_BIG), 256, 0, stream>>>(bufP, SZ_BIG);
  k_prop<<<blks((long)ESSc * Bc * Tc), 256, 0, stream>>>(eind_ss, nrmS, ESSc, Nc,
                                                         bufH, bufP);
  k_cheb<<<blks(SZ_BTN), 256, 0, stream>>>(bufG, bufH, bufP, wc_s, nullptr, bufG,
                                           SZ_BTN);

  // 4) temporal attention (both halves): bufG -> bufH
  k_ta<<<(unsigned)(Bc * Nc * 2), 96, 0, stream>>>(bufG, wq_t, wk_t, wv_t, bufH);

  // 5) gate + scale+relu (in place on bufH)
  k_gate<<<Bc, 256, 0, stream>>>(bufH, gw_q, gw_f, gw_n, gbuf);
  k_scale<<<blks(SZ_BIG), 256, 0, stream>>>(bufH, gbuf);

  // 6) r = relu(scaled @ adj_sr) (WMMA) -> bufR
  {
    int ntiles = (NRc + 15) >> 4;  // 7
    k_rmat<<<(unsigned)(Bc * Tc * ntiles), 32, 0, stream>>>(adj_sr, bufH, bufR);
  }

  // 7) GAT-R projections + GAT-R (WMMA) -> gatR
  k_proj<<<blks(SZ_BTR), 256, 0, stream>>>(bufR, wg_r, a1_r, a2_r, hR, alR, beR,
                                           SZ_BTR);
  {
    int ntiles = (NRc + 15) >> 4;
    k_gat<<<(unsigned)(Bc * Tc * ntiles), 32, 0, stream>>>(adj_rr, NRc, hR, alR,
                                                           beR, gatR);
  }

  // 8) Cheb-R
  k_zero<<<blks(NRc), 256, 0, stream>>>(degR, NRc);
  k_deg<<<blks(ERRc), 256, 0, stream>>>(eind_rr, ERRc, degR);
  k_norm<<<blks(ERRc), 256, 0, stream>>>(eind_rr, ERRc, degR, nrmR);
  k_zero<<<blks(SZ_R), 256, 0, stream>>>(tx1R, SZ_R);
  k_prop<<<blks((long)ERRc * Bc * Tc), 256, 0, stream>>>(eind_rr, nrmR, ERRc, NRc,
                                                         gatR, tx1R);
  k_zero<<<blks(SZ_R), 256, 0, stream>>>(prR, SZ_R);
  k_prop<<<blks((long)ERRc * Bc * Tc), 256, 0, stream>>>(eind_rr, nrmR, ERRc, NRc,
                                                         tx1R, prR);
  k_cheb<<<blks(SZ_BTR), 256, 0, stream>>>(gatR, tx1R, prR, wc_r, b_chebr, gatR,
                                           SZ_BTR);

  // 9) output contraction -> d_out [B, DIN]
  k_out<<<(unsigned)(Bc * DINc), 256, 0, stream>>>(gatR, w_out, b_out, outp);
}